// CausalNet_77936476553901
// MI455X (gfx1250) — compile-verified
//
#include <hip/hip_runtime.h>
#include <stdint.h>

#define B_   32
#define T_   256
#define NB_  16
#define NEG_ (-1e9f)

// ---------------- wave32 reductions (gfx1250 is wave32) ----------------
__device__ __forceinline__ float wave_max32(float v) {
#pragma unroll
  for (int o = 16; o > 0; o >>= 1) v = fmaxf(v, __shfl_xor(v, o, 32));
  return v;
}
__device__ __forceinline__ float wave_sum32(float v) {
#pragma unroll
  for (int o = 16; o > 0; o >>= 1) v += __shfl_xor(v, o, 32);
  return v;
}

// ------------- CDNA5 async global->LDS copy (ASYNCcnt path) -------------
__device__ __forceinline__ void async_ld_b128(uint32_t lds_off, const void* g) {
  asm volatile("global_load_async_to_lds_b128 %0, %1, off"
               :: "v"(lds_off), "v"((uint64_t)(size_t)g)
               : "memory");
}
#define WAIT_ASYNCCNT(n) asm volatile("s_wait_asynccnt " #n ::: "memory")

__device__ __forceinline__ uint32_t lds_off_of(const void* p) {
  // generic LDS pointer = {SHARED_BASE, offset[31:0]}; low 32 bits = LDS offset
  return (uint32_t)(size_t)p;
}

// ---------------- Tensor Data Mover (TENSORcnt path) ----------------
// Loads a 2D tile: tile_dim0=16 elems (contiguous, 64B), tile_dim1=256 rows,
// row stride 4112 elements (257*16 floats) -> 16KB dense in LDS.
typedef unsigned int u32x4_t __attribute__((ext_vector_type(4)));
typedef int          i32x4_t __attribute__((ext_vector_type(4)));
typedef int          i32x8_t __attribute__((ext_vector_type(8)));

__device__ __forceinline__ void tdm_load_cp_tile(uint32_t lds_addr, uint64_t gaddr) {
  u32x4_t g0;
  g0.x = 1u;                                   // count=1 (valid user descriptor)
  g0.y = lds_addr;                             // D# group0[63:32]  = lds_addr
  g0.z = (uint32_t)gaddr;                      // global_addr[31:0]
  g0.w = (uint32_t)((gaddr >> 32) & 0x01FFFFFFull) | (2u << 30); // addr[56:32]|type=2
  i32x8_t g1;
  g1.s0 = 0x00020000;  // workgroup_mask=0, data_size=2 (4B), no pad/iterate
  g1.s1 = 16 << 16;    // tensor_dim0[15:0]=16  (bits 63:48)
  g1.s2 = 256 << 16;   // tensor_dim0 hi=0, tensor_dim1[15:0]=256 (bits 95:80)
  g1.s3 = 16 << 16;    // tensor_dim1 hi=0, tile_dim0=16 (bits 127:112)
  g1.s4 = 256;         // tile_dim1=256, tile_dim2=0
  g1.s5 = 4112;        // tensor_dim0_stride[31:0] = 257*16 elements
  g1.s6 = 0;           // stride hi / tensor_dim1_stride lo
  g1.s7 = 0;
  i32x4_t z4 = {0, 0, 0, 0};                   // groups 2/3: tile_dim2..4 unused
  i32x8_t z8 = {0, 0, 0, 0, 0, 0, 0, 0};       // trailing group (clang-23 form)
  __builtin_amdgcn_tensor_load_to_lds(g0, g1, z4, z4, z8, 0);
}

// ======================= backward scan (cc_bw) =======================
// 1 wave per batch element. lane = e*16 + j. Emits BW[b][tau] = bw1 scalar
// and TLs[b] = total_logp at t=0 (used only as a numeric shift).
__global__ __launch_bounds__(32) void cc_bw_kernel(
    const float* __restrict__ act, const float* __restrict__ stp,
    const float* __restrict__ str, const int* __restrict__ masks,
    float* __restrict__ BW, float* __restrict__ TLs)
{
  const int b = blockIdx.x;
  const int lane = threadIdx.x;
  const int j = lane & 15;
  const int e = lane >> 4;

  float g = (e == 1) ? 0.0f : NEG_;                     // g_init = [NEG, 0]
  if (lane == 0) BW[(size_t)b * T_ + (T_ - 1)] = 0.0f;  // bw1[T-1] = 0

  for (int t = T_ - 1; t >= 0; --t) {
    const float a  = act[((size_t)b * T_ + t) * NB_ + j];
    const float sp = stp[(((size_t)b * 257 + (t + 1)) * NB_ + j) * 2 + e];
    const float st = str[((size_t)b * 257 + t) * NB_ + j];

    const float x  = sp + g;
    const float xp = __shfl_xor(x, 16, 32);             // partner e-channel
    const float m2 = fmaxf(x, xp);
    const float lse2 = m2 + __logf(1.0f + __expf(-fabsf(x - xp)));
    const float g0 = a + lse2;

    const float y = st + a + x;
    const float M = wave_max32(y);
    const float S = wave_sum32(__expf(y - M));
    const float g1 = M + __logf(S);

    const int m = masks[(size_t)b * T_ + t];
    const float g0m = (m == 1) ? g0 : NEG_;
    const float g1m = (m == 1) ? g1 : 0.0f;
    g = (e == 0) ? g0m : g1m;

    if (lane == 0) {
      if (t >= 1) BW[(size_t)b * T_ + (t - 1)] = g1m;   // bw1[t-1]
      else        TLs[b] = g1;                           // fw/bw identity
    }
  }
}

// ================= forward scan + causal-consistency =================
// 1 block (256 thr) per batch. thread tid owns start-slot s=tid, holds the
// e=0 lattice channel for all 16 options in VGPRs. causal_pens tiles
// (16KB/step) are TDM-loaded into an LDS triple buffer, depth-2 pipeline.
__global__ __launch_bounds__(256) void cc_fw_kernel(
    const float* __restrict__ act, const float* __restrict__ stp,
    const float* __restrict__ str, const float* __restrict__ cp,
    const int* __restrict__ masks, const int* __restrict__ lengths,
    const float* __restrict__ BW, const float* __restrict__ TLs,
    float* __restrict__ partials)
{
  __shared__ __align__(16) float lds_cp[3][T_ * NB_];  // 48KB triple buffer
  __shared__ __align__(16) float lds_bw[T_];           // 1KB
  __shared__ float red_m[8];
  __shared__ float red_s[8];

  const int b   = blockIdx.x;
  const int tid = threadIdx.x;           // start-slot s
  const int wid = tid >> 5, lane = tid & 31;

  const uint64_t cp_base = (uint64_t)(size_t)(cp + ((size_t)b * 257) * 257 * NB_);
  const uint32_t lds_buf[3] = { lds_off_of(&lds_cp[0][0]),
                                lds_off_of(&lds_cp[1][0]),
                                lds_off_of(&lds_cp[2][0]) };

  // ---- prologue: async-preload BW row; TDM-prime tiles t=1, t=2 ----
  if (tid < 64)
    async_ld_b128(lds_off_of(&lds_bw[tid * 4]),
                  (const char*)(BW + (size_t)b * T_) + tid * 16);
  WAIT_ASYNCCNT(0);
  if (wid == 0) {
    tdm_load_cp_tile(lds_buf[1 % 3], cp_base + (uint64_t)1 * NB_ * 4);
    tdm_load_cp_tile(lds_buf[2 % 3], cp_base + (uint64_t)2 * NB_ * 4);
  }
  __syncthreads();

  float f[NB_];                          // f[t][j, s=tid, e=0]
#pragma unroll
  for (int j = 0; j < NB_; ++j) f[j] = NEG_;
  float psl = 0.0f;                      // prev_stop_lse(f0) = 0
  float acc = 0.0f;                      // cc partial (shift-normalized)
  float TLfw = NEG_;
  const float tls = TLs[b];
  const int len = lengths[b];

  for (int t = 1; t <= T_; ++t) {
    // wave0: prefetch tile t+2 (depth-2 pipeline), then retire tile t.
    // TENSORcnt retires in issue order, so the wait threshold equals the
    // number of younger tiles still allowed in flight.
    if (wid == 0) {
      if (t + 2 <= T_) {
        tdm_load_cp_tile(lds_buf[(t + 2) % 3],
                         cp_base + (uint64_t)(t + 2) * NB_ * 4);
        __builtin_amdgcn_s_wait_tensorcnt(2);
      } else if (t + 1 <= T_) {
        __builtin_amdgcn_s_wait_tensorcnt(1);
      } else {
        __builtin_amdgcn_s_wait_tensorcnt(0);
      }
    }
    __syncthreads();

    // per-step params: wave-uniform addresses -> scalar (SMEM) loads
    const float* aR  = act + ((size_t)b * T_  + (t - 1)) * NB_;
    const float* spR = stp + ((size_t)b * 257 + t)       * (NB_ * 2);
    const float* stR = str + ((size_t)b * 257 + (t - 1)) * NB_;

    float f1[NB_];
    float lmax = NEG_;
#pragma unroll
    for (int j = 0; j < NB_; ++j) {
      const float a  = aR[j];
      const float c0 = spR[2 * j];       // stop[...,continue]
      const float c1 = spR[2 * j + 1];   // stop[...,stop]
      if (tid < t - 1)       f[j] += a + c0;                  // continue
      else if (tid == t - 1) f[j]  = psl + stR[j] + a + c0;   // fresh entry
      const float v = f[j] + (c1 - c0);  // e=1 channel
      f1[j] = v;
      if (tid < t) lmax = fmaxf(lmax, v);
    }

    // block max for a stable lse
    const float wm = wave_max32(lmax);
    if (lane == 0) red_m[wid] = wm;
    __syncthreads();
    float M = red_m[0];
#pragma unroll
    for (int w = 1; w < 8; ++w) M = fmaxf(M, red_m[w]);

    // one exp per element feeds both the lse sum and the cc dot
    const float* cpv = &lds_cp[t % 3][tid * NB_];
    float se = 0.0f, dot = 0.0f;
#pragma unroll
    for (int j = 0; j < NB_; ++j) {
      const float e = __expf(f1[j] - M); // tid>=t: f1 ~ -1e9 -> 0 (triangle)
      se  += e;
      dot += e * cpv[j];
    }
    if (masks[(size_t)b * T_ + (t - 1)] == 1)
      acc += dot * __expf(M + lds_bw[t - 1] - tls);

    const float ws = wave_sum32(se);
    if (lane == 0) red_s[wid] = ws;
    __syncthreads();
    float S = 0.0f;
#pragma unroll
    for (int w = 0; w < 8; ++w) S += red_s[w];
    psl = M + __logf(S);                 // prev_stop_lse for step t+1
    if (t == len) TLfw = psl;            // total_logp (reference path: fw lse)
  }

  // block-reduce cc partials; renormalize shift to forward total_logp
  __syncthreads();
  const float wa = wave_sum32(acc);
  if (lane == 0) red_s[wid] = wa;
  __syncthreads();
  if (tid == 0) {
    float cc = 0.0f;
#pragma unroll
    for (int w = 0; w < 8; ++w) cc += red_s[w];
    // CC_WEIGHT = 1.0
    partials[b] = -TLfw + cc * __expf(tls - TLfw);
  }
}

// fixed-order final sum -> deterministic scalar output
__global__ void cc_fin_kernel(const float* __restrict__ partials,
                              float* __restrict__ out)
{
  if (blockIdx.x == 0 && threadIdx.x == 0) {
    float s = 0.0f;
    for (int i = 0; i < B_; ++i) s += partials[i];
    out[0] = s;
  }
}

extern "C" void kernel_launch(void* const* d_in, const int* in_sizes, int n_in,
                              void* d_out, int out_size, void* d_ws, size_t ws_size,
                              hipStream_t stream) {
  (void)in_sizes; (void)n_in; (void)out_size; (void)ws_size;
  const float* act     = (const float*)d_in[0];  // [B, T, NB]
  const float* stp     = (const float*)d_in[1];  // [B, T+1, NB, 2]
  const float* str     = (const float*)d_in[2];  // [B, T+1, NB]
  const float* cpens   = (const float*)d_in[3];  // [B, T+1, T+1, NB]
  const int*   lengths = (const int*)d_in[4];    // [B]
  const int*   masks   = (const int*)d_in[5];    // [B, T]

  float* BW   = (float*)d_ws;            // B*T floats (32KB)
  float* TLs  = BW + (size_t)B_ * T_;    // B floats
  float* part = TLs + B_;                // B floats

  cc_bw_kernel<<<B_, 32, 0, stream>>>(act, stp, str, masks, BW, TLs);
  cc_fw_kernel<<<B_, 256, 0, stream>>>(act, stp, str, cpens, masks, lengths,
                                       BW, TLs, part);
  cc_fin_kernel<<<1, 32, 0, stream>>>(part, (float*)d_out);
}